// RNN_71983651881619
// MI455X (gfx1250) — compile-verified
//
#include <hip/hip_runtime.h>
#include <stdint.h>

// GRU (I=2, H=2) over T=512, B=32768, fused with exp(fc(h)).
// Memory-bound streaming recurrence: ~200MB traffic -> ~9us roofline @23.3TB/s.
// One thread per batch element; serial loop over T; weights in SGPRs.
// CDNA5 paths: global_load_async_to_lds_b64 (ASYNCcnt) 32-deep pipeline with
// s_wait_asynccnt gating, non-temporal hints on the read-once/write-once streams.
// WMMA intentionally unused: the matmuls are 2x2 (<=2% MAC utilization) and the
// D->A relayout would need cross-lane permutes inside the serial T chain.

#define TPB    64
#define DEPTH  32          // outstanding async loads per wave (ASYNCcnt is 6-bit)
#define SLOTS  64          // ring slots (2x depth -> no read/refill WAR on a slot)

__device__ __forceinline__ float fexp(float v) {
#if __has_builtin(__builtin_amdgcn_exp2f)
  return __builtin_amdgcn_exp2f(v * 1.44269504088896340736f);  // v_exp_f32
#else
  return __expf(v);
#endif
}
__device__ __forceinline__ float frcp(float v) {
#if __has_builtin(__builtin_amdgcn_rcpf)
  return __builtin_amdgcn_rcpf(v);                              // v_rcp_f32
#else
  return 1.0f / v;
#endif
}
__device__ __forceinline__ float fsigmoid(float v) { return frcp(1.0f + fexp(-v)); }
__device__ __forceinline__ float ftanh(float v) {
#if __has_builtin(__builtin_amdgcn_tanhf)
  return __builtin_amdgcn_tanhf(v);                             // v_tanh_f32 (gfx1250)
#else
  float e = fexp(-2.0f * v);
  return (1.0f - e) * frcp(1.0f + e);
#endif
}

__global__ __launch_bounds__(TPB) void gru_fused(
    const float* __restrict__ x,    const float* __restrict__ hx,
    const float* __restrict__ W_ih, const float* __restrict__ W_hh,
    const float* __restrict__ b_ih, const float* __restrict__ b_hh,
    const float* __restrict__ fc_w, const float* __restrict__ fc_b,
    float* __restrict__ out, float* __restrict__ h_out, int T, int B)
{
  __shared__ float xs[SLOTS][TPB][2];   // 64*64*2*4 = 32KB ring buffer

  const int tid = threadIdx.x;
  const int b   = blockIdx.x * TPB + tid;
  if (b >= B) return;

  // ---- weights -> registers (uniform addresses -> scalar loads/SGPRs) ----
  float wih[6][2], whh[6][2], bi[6], bh[6];
#pragma unroll
  for (int g = 0; g < 6; ++g) {
    wih[g][0] = W_ih[g * 2 + 0]; wih[g][1] = W_ih[g * 2 + 1];
    whh[g][0] = W_hh[g * 2 + 0]; whh[g][1] = W_hh[g * 2 + 1];
    bi[g] = b_ih[g]; bh[g] = b_hh[g];
  }
  const float fw0 = fc_w[0], fw1 = fc_w[1], fb = fc_b[0];

  float h0 = hx[(size_t)b * 2 + 0];
  float h1 = hx[(size_t)b * 2 + 1];

  // ---- async-to-LDS pipeline setup ----
  // LDS byte offset of this thread's lane slot base (low 32 bits of flat LDS addr).
  const uint32_t lds_base  = (uint32_t)(uintptr_t)(&xs[0][0][0]) + (uint32_t)tid * 8u;
  const uint64_t gbase     = (uint64_t)(uintptr_t)x + (uint64_t)b * 8u;   // &x[0][b][0]
  const uint64_t row_bytes = (uint64_t)B * 8u;                            // t -> t+1
  uint64_t gaddr = gbase;

  // Prologue: always issue exactly DEPTH async loads (dummy in-range loads if T<DEPTH)
  // so the steady-state wait threshold (DEPTH-1) is uniformly correct.
  // x is read exactly once -> non-temporal: don't churn L2 with the 134MB stream.
#pragma unroll 4
  for (int t = 0; t < DEPTH; ++t) {
    const uint64_t a = (t < T) ? gaddr : gbase;
    asm volatile("global_load_async_to_lds_b64 %0, %1, off th:TH_LOAD_NT"
                 :: "v"(lds_base + (uint32_t)(t * (TPB * 8))), "v"(a) : "memory");
    gaddr += row_bytes;
  }
  // gaddr now points at row t = DEPTH.

  float* op = out + b;
  for (int t = 0; t < T; ++t) {
    const int rslot = t & (SLOTS - 1);

    // Oldest of the DEPTH outstanding loads (slot rslot) has landed in LDS.
    asm volatile("s_wait_asynccnt %0" :: "n"(DEPTH - 1) : "memory");

    const float x0 = xs[rslot][tid][0];
    const float x1 = xs[rslot][tid][1];

    // Refill for step t+DEPTH into slot (t+DEPTH)&63 (!= rslot: 2x ring, so the
    // in-flight ds_load and the new async write never touch the same slot).
    // Past the end, issue a dummy load of row 0 into a slot never read again.
    {
      const int      wslot = (t + DEPTH) & (SLOTS - 1);
      const uint64_t a     = (t + DEPTH < T) ? gaddr : gbase;
      asm volatile("global_load_async_to_lds_b64 %0, %1, off th:TH_LOAD_NT"
                   :: "v"(lds_base + (uint32_t)(wslot * (TPB * 8))), "v"(a) : "memory");
      gaddr += row_bytes;
    }

    // ---- GRU cell (PyTorch gate order r,z,n; H=2) ----
    const float gi0 = fmaf(x1, wih[0][1], fmaf(x0, wih[0][0], bi[0]));
    const float gi1 = fmaf(x1, wih[1][1], fmaf(x0, wih[1][0], bi[1]));
    const float gi2 = fmaf(x1, wih[2][1], fmaf(x0, wih[2][0], bi[2]));
    const float gi3 = fmaf(x1, wih[3][1], fmaf(x0, wih[3][0], bi[3]));
    const float gi4 = fmaf(x1, wih[4][1], fmaf(x0, wih[4][0], bi[4]));
    const float gi5 = fmaf(x1, wih[5][1], fmaf(x0, wih[5][0], bi[5]));

    const float gh0 = fmaf(h1, whh[0][1], fmaf(h0, whh[0][0], bh[0]));
    const float gh1 = fmaf(h1, whh[1][1], fmaf(h0, whh[1][0], bh[1]));
    const float gh2 = fmaf(h1, whh[2][1], fmaf(h0, whh[2][0], bh[2]));
    const float gh3 = fmaf(h1, whh[3][1], fmaf(h0, whh[3][0], bh[3]));
    const float gh4 = fmaf(h1, whh[4][1], fmaf(h0, whh[4][0], bh[4]));
    const float gh5 = fmaf(h1, whh[5][1], fmaf(h0, whh[5][0], bh[5]));

    const float r0 = fsigmoid(gi0 + gh0);
    const float r1 = fsigmoid(gi1 + gh1);
    const float z0 = fsigmoid(gi2 + gh2);
    const float z1 = fsigmoid(gi3 + gh3);
    const float n0 = ftanh(fmaf(r0, gh4, gi4));
    const float n1 = ftanh(fmaf(r1, gh5, gi5));

    h0 = fmaf(z0, h0 - n0, n0);   // (1-z)*n + z*h
    h1 = fmaf(z1, h1 - n1, n1);

    // fused head: exp(fc(h)) ; streaming write-once -> non-temporal store
    const float o = fexp(fmaf(fw1, h1, fmaf(fw0, h0, fb)));
    __builtin_nontemporal_store(o, op);
    op += B;
  }

  // h_last: [1, B, 2]
  h_out[(size_t)b * 2 + 0] = h0;
  h_out[(size_t)b * 2 + 1] = h1;
}

extern "C" void kernel_launch(void* const* d_in, const int* in_sizes, int n_in,
                              void* d_out, int out_size, void* d_ws, size_t ws_size,
                              hipStream_t stream) {
  const float* x    = (const float*)d_in[0];
  const float* hx   = (const float*)d_in[1];
  const float* W_ih = (const float*)d_in[2];
  const float* W_hh = (const float*)d_in[3];
  const float* b_ih = (const float*)d_in[4];
  const float* b_hh = (const float*)d_in[5];
  const float* fc_w = (const float*)d_in[6];
  const float* fc_b = (const float*)d_in[7];

  const int B = in_sizes[1] / 2;            // hx is [1,B,2]
  const int T = in_sizes[0] / (B * 2);      // x  is [T,B,2]

  float* out   = (float*)d_out;             // [T,B,1] flat
  float* h_out = out + (size_t)T * B;       // [1,B,2] flat, concatenated after

  const int grid = (B + TPB - 1) / TPB;     // 512 blocks -> spread across WGPs
  gru_fused<<<grid, TPB, 0, stream>>>(x, hx, W_ih, W_hh, b_ih, b_hh,
                                      fc_w, fc_b, out, h_out, T, B);
}